// SoftDotAttention_52604759441990
// MI455X (gfx1250) — compile-verified
//
#include <hip/hip_runtime.h>
#include <hip/hip_bf16.h>

// SoftDotAttention for MI455X (gfx1250, wave32).
//   B=128, S=1024, D=1024, fp32 end-to-end.
// d_in: [0]=h[B,D] f32, [1]=context[B,S,D] f32, [2]=mask[B,S] bool(1B),
//       [3]=W_in[D,D] f32, [4]=W_out[D,2D] f32
// d_out: h_tilde[B,D] f32 followed by attn[B,S] f32
// d_ws:  target[B,D] f32 then wc[B,D] f32  (1 MB)

#define BB 128
#define SS 1024
#define DD 1024

typedef __attribute__((ext_vector_type(2))) float v2f;
typedef __attribute__((ext_vector_type(8))) float v8f;

// ---------------------------------------------------------------------------
// Kernel 1: target = h @ W_in^T   (M=128, N=1024, K=1024), fp32 WMMA 16x16x4.
// One wave owns one 16x16 output tile. A operand (16x4 fp32, 2 VGPRs):
//   lanes 0-15 hold K=k..k+1 of row m0+lane, lanes 16-31 hold K=k+2..k+3.
// B operand (4x16 fp32, 2 VGPRs) mirrors that with rows of W_in (B = W_in^T).
// ---------------------------------------------------------------------------
__global__ __launch_bounds__(128) void sda_gemm_target(
    const float* __restrict__ h, const float* __restrict__ Win,
    float* __restrict__ target) {
  const int lane = threadIdx.x & 31;
  const int wave = threadIdx.x >> 5;
  const int m0 = blockIdx.x * 16;
  const int n0 = (blockIdx.y * 4 + wave) * 16;
  const int r = lane & 15;
  const int half = lane >> 4;      // 0: lanes 0-15, 1: lanes 16-31
  const int koff = half * 2;       // K sub-offset per documented layout

  const float* arow = h + (size_t)(m0 + r) * DD;   // row m0+r of h
  const float* brow = Win + (size_t)(n0 + r) * DD; // row n0+r of W_in

  v8f acc = {};
#pragma unroll 4
  for (int k = 0; k < DD; k += 4) {
    v2f a = *(const v2f*)(arow + k + koff);  // A[m, k+koff .. k+koff+1]
    v2f b = *(const v2f*)(brow + k + koff);  // B[k+koff.., n] = W_in[n, k+koff..]
    acc = __builtin_amdgcn_wmma_f32_16x16x4_f32(
        /*neg_a=*/false, a, /*neg_b=*/false, b,
        /*c_mod=*/(short)0, acc, /*reuse_a=*/false, /*reuse_b=*/false);
  }

  // C/D layout: VGPR i -> M = m0 + i + 8*half, N = n0 + r
  float* out = target + (size_t)(m0 + 8 * half) * DD + n0 + r;
#pragma unroll
  for (int i = 0; i < 8; ++i) out[(size_t)i * DD] = acc[i];
}

// ---------------------------------------------------------------------------
// Wave reductions (wave32).
// ---------------------------------------------------------------------------
__device__ inline float wave_sum(float v) {
#pragma unroll
  for (int off = 16; off > 0; off >>= 1) v += __shfl_xor(v, off, 32);
  return v;
}
__device__ inline float wave_max(float v) {
#pragma unroll
  for (int off = 16; off > 0; off >>= 1) v = fmaxf(v, __shfl_xor(v, off, 32));
  return v;
}

// ---------------------------------------------------------------------------
// Kernel 2: fused scores -> masked softmax -> weighted context, one block per
// batch. Both context passes for a batch happen back-to-back so the 4 MB slice
// stays L2-resident (192 MB L2) for the second pass.
// 256 threads = 8 waves.
// ---------------------------------------------------------------------------
__global__ __launch_bounds__(256) void sda_attn_fused(
    const float* __restrict__ ctx, const float* __restrict__ target,
    const unsigned char* __restrict__ mask, float* __restrict__ attn_out,
    float* __restrict__ wc) {
  const int b = blockIdx.x;
  const int tid = threadIdx.x;
  const int lane = tid & 31;
  const int wave = tid >> 5;

  __shared__ float tgt[DD];   // target[b]
  __shared__ float sc[SS];    // scores -> probabilities
  __shared__ float red[8];

  for (int i = tid; i < DD; i += 256) tgt[i] = target[(size_t)b * DD + i];
  __syncthreads();

  const float* cbase = ctx + (size_t)b * SS * DD;

  // Phase 1: scores. Each wave computes one row's dot product cooperatively:
  // lane l streams float4 at k = l*4 + j*128 (512B coalesced per wave step).
  for (int s = wave; s < SS; s += 8) {
    const float* row = cbase + (size_t)s * DD;
    float acc = 0.f;
#pragma unroll
    for (int k = lane * 4; k < DD; k += 128) {
      float4 c4 = *(const float4*)(row + k);
      acc += c4.x * tgt[k] + c4.y * tgt[k + 1] + c4.z * tgt[k + 2] +
             c4.w * tgt[k + 3];
    }
    acc = wave_sum(acc);
    if (lane == 0)
      sc[s] = mask[(size_t)b * SS + s] ? -__builtin_inff() : acc;
  }
  __syncthreads();

  // Phase 2: softmax over S=1024 with 256 threads.
  float lmax = -__builtin_inff();
  for (int i = tid; i < SS; i += 256) lmax = fmaxf(lmax, sc[i]);
  lmax = wave_max(lmax);
  if (lane == 0) red[wave] = lmax;
  __syncthreads();
  float gmax = red[0];
#pragma unroll
  for (int i = 1; i < 8; ++i) gmax = fmaxf(gmax, red[i]);
  __syncthreads();

  float lsum = 0.f;
  for (int i = tid; i < SS; i += 256) {
    float e = __expf(sc[i] - gmax);
    sc[i] = e;
    lsum += e;
  }
  lsum = wave_sum(lsum);
  __syncthreads();
  if (lane == 0) red[wave] = lsum;
  __syncthreads();
  float gsum = 0.f;
#pragma unroll
  for (int i = 0; i < 8; ++i) gsum += red[i];
  const float inv = 1.0f / gsum;  // all-masked row -> inf/NaN like reference

  for (int i = tid; i < SS; i += 256) {
    float p = sc[i] * inv;
    sc[i] = p;
    attn_out[(size_t)b * SS + i] = p;
  }
  __syncthreads();

  // Phase 3: wc[b, :] = sum_s p[s] * ctx[b, s, :].
  // Thread owns 4 consecutive d (float4): block covers all 1024 d, loads are
  // fully coalesced 4 KB rows; rows should hit L2 from phase 1.
  const float* col = cbase + tid * 4;
  float4 acc = {0.f, 0.f, 0.f, 0.f};
  for (int s = 0; s < SS; s += 4) {
    float p0 = sc[s], p1 = sc[s + 1], p2 = sc[s + 2], p3 = sc[s + 3];
    float4 c0 = *(const float4*)(col + (size_t)(s + 0) * DD);
    float4 c1 = *(const float4*)(col + (size_t)(s + 1) * DD);
    float4 c2 = *(const float4*)(col + (size_t)(s + 2) * DD);
    float4 c3 = *(const float4*)(col + (size_t)(s + 3) * DD);
    acc.x += p0 * c0.x + p1 * c1.x + p2 * c2.x + p3 * c3.x;
    acc.y += p0 * c0.y + p1 * c1.y + p2 * c2.y + p3 * c3.y;
    acc.z += p0 * c0.z + p1 * c1.z + p2 * c2.z + p3 * c3.z;
    acc.w += p0 * c0.w + p1 * c1.w + p2 * c2.w + p3 * c3.w;
  }
  *(float4*)(wc + (size_t)b * DD + tid * 4) = acc;
}

// ---------------------------------------------------------------------------
// Kernel 3: h_tilde = tanh( [wc | h] @ W_out^T )  (M=128, N=1024, K=2048).
// Same WMMA tiling; K split into two uniform loops (wc half, h half).
// ---------------------------------------------------------------------------
__global__ __launch_bounds__(128) void sda_gemm_out(
    const float* __restrict__ wc, const float* __restrict__ h,
    const float* __restrict__ Wout, float* __restrict__ htilde) {
  const int lane = threadIdx.x & 31;
  const int wave = threadIdx.x >> 5;
  const int m0 = blockIdx.x * 16;
  const int n0 = (blockIdx.y * 4 + wave) * 16;
  const int r = lane & 15;
  const int half = lane >> 4;
  const int koff = half * 2;

  const float* arow0 = wc + (size_t)(m0 + r) * DD;       // k in [0, 1024)
  const float* arow1 = h + (size_t)(m0 + r) * DD;        // k in [1024, 2048)
  const float* brow = Wout + (size_t)(n0 + r) * (2 * DD);  // W_out row, stride 2D

  v8f acc = {};
#pragma unroll 4
  for (int k = 0; k < DD; k += 4) {
    v2f a = *(const v2f*)(arow0 + k + koff);
    v2f b = *(const v2f*)(brow + k + koff);
    acc = __builtin_amdgcn_wmma_f32_16x16x4_f32(false, a, false, b, (short)0,
                                                acc, false, false);
  }
#pragma unroll 4
  for (int k = 0; k < DD; k += 4) {
    v2f a = *(const v2f*)(arow1 + k + koff);
    v2f b = *(const v2f*)(brow + DD + k + koff);
    acc = __builtin_amdgcn_wmma_f32_16x16x4_f32(false, a, false, b, (short)0,
                                                acc, false, false);
  }

  float* out = htilde + (size_t)(m0 + 8 * half) * DD + n0 + r;
#pragma unroll
  for (int i = 0; i < 8; ++i) out[(size_t)i * DD] = tanhf(acc[i]);
}

// ---------------------------------------------------------------------------
extern "C" void kernel_launch(void* const* d_in, const int* in_sizes, int n_in,
                              void* d_out, int out_size, void* d_ws,
                              size_t ws_size, hipStream_t stream) {
  const float* h = (const float*)d_in[0];
  const float* ctx = (const float*)d_in[1];
  const unsigned char* mask = (const unsigned char*)d_in[2];  // numpy bool_
  const float* Win = (const float*)d_in[3];
  const float* Wout = (const float*)d_in[4];

  float* htilde = (float*)d_out;                       // [B, D]
  float* attn = (float*)d_out + (size_t)BB * DD;       // [B, S]

  float* target = (float*)d_ws;                        // [B, D]
  float* wc = target + (size_t)BB * DD;                // [B, D]

  // K1: target = h @ W_in^T
  sda_gemm_target<<<dim3(BB / 16, DD / 64), 128, 0, stream>>>(h, Win, target);
  // K2: fused masked-softmax attention (attn + weighted context), 1 block/batch
  sda_attn_fused<<<BB, 256, 0, stream>>>(ctx, target, mask, attn, wc);
  // K3: h_tilde = tanh([wc|h] @ W_out^T)
  sda_gemm_out<<<dim3(BB / 16, DD / 64), 128, 0, stream>>>(wc, h, Wout, htilde);
}